// ARboxMultiRoIExtractor_22874995819151
// MI455X (gfx1250) — compile-verified
//
#include <hip/hip_runtime.h>
#include <hip/hip_bf16.h>
#include <cstdint>

#define OUT_H 7
#define OUT_W 35
#define NBINS (OUT_H * OUT_W)
#define CTOT 256
#define CCHUNK 16

// One block = (one roi) x (16-channel chunk). Threads = (35,7): one bin each,
// lanes walk along ox so wave gathers hit consecutive x (few 128B lines/req).
__global__ __launch_bounds__(NBINS) void rroi_align_max_kernel(
    const float* __restrict__ f0, const float* __restrict__ f1,
    const float* __restrict__ f2, const float* __restrict__ f3,
    const float* __restrict__ rois, float* __restrict__ out)
{
    const int n  = blockIdx.x;
    const int c0 = blockIdx.y * CCHUNK;
    const int ox = threadIdx.x;
    const int oy = threadIdx.y;
    const int tid = oy * OUT_W + ox;          // == bin index

    // ---- stage roi row (24B) to LDS via gfx1250 async global->LDS path ----
    __shared__ float s_roi[8];
    if (tid < 6) {
        unsigned lds_off = (unsigned)(uintptr_t)&s_roi[tid];   // low 32b of generic
        const float* ga  = rois + (size_t)n * 6 + tid;
        asm volatile("global_load_async_to_lds_b32 %0, %1, off"
                     :: "v"(lds_off), "v"(ga) : "memory");
    }
    asm volatile("s_wait_asynccnt 0" ::: "memory");
    __syncthreads();

    const int   bIdx  = (int)s_roi[0];
    const float cx    = s_roi[1];
    const float cy    = s_roi[2];
    const float rw    = s_roi[3];
    const float rh    = s_roi[4];
    const float theta = s_roi[5];
    const float ct = cosf(theta), st = sinf(theta);

    const float* feats[4] = { f0, f1, f2, f3 };
    const int    HS[4]    = { 256, 128, 64, 32 };
    const float  SC[4]    = { 0.25f, 0.125f, 0.0625f, 0.03125f };

    float acc[CCHUNK];
#pragma unroll
    for (int i = 0; i < CCHUNK; ++i) acc[i] = -3.402823466e38f;

    for (int l = 0; l < 4; ++l) {
        const int   H  = HS[l];
        const float sc = SC[l];
        const float Hf = (float)H;

        const float wpx = fmaxf(rw * sc, 1.0f);
        const float hpx = fmaxf(rh * sc, 1.0f);
        const float bw  = wpx * (1.0f / OUT_W);
        const float bh  = hpx * (1.0f / OUT_H);
        const float cxs = cx * sc, cys = cy * sc;

        // Per sample: base offset of a contiguous 2x2 patch + 4 weights.
        // Edge clamp folded as xb=min(x0,H-2): interior -> identical bilinear,
        // boundary -> weights (0,1) on the pair == reference's duplicated corner.
        int   off[4];       // yb*H + xb
        float wt[4][4];     // w00,w01,w10,w11 (pre-multiplied by valid*0.25)
#pragma unroll
        for (int s = 0; s < 4; ++s) {
            const int sy = s >> 1, sx = s & 1;
            const float gy = (float)oy + ((float)sy + 0.5f) * 0.5f;
            const float gx = (float)ox + ((float)sx + 0.5f) * 0.5f;
            const float yy = gy * bh - 0.5f * hpx;
            const float xx = gx * bw - 0.5f * wpx;
            float xf = xx * ct - yy * st + cxs;
            float yf = xx * st + yy * ct + cys;
            const float vmul =
                (yf >= -1.0f && yf <= Hf && xf >= -1.0f && xf <= Hf) ? 0.25f : 0.0f;
            yf = fminf(fmaxf(yf, 0.0f), Hf - 1.0f);
            xf = fminf(fmaxf(xf, 0.0f), Hf - 1.0f);
            const int yb = min((int)floorf(yf), H - 2);
            const int xb = min((int)floorf(xf), H - 2);
            const float ly = yf - (float)yb, lx = xf - (float)xb;
            const float hy = 1.0f - ly,      hx = 1.0f - lx;
            off[s]   = yb * H + xb;
            wt[s][0] = vmul * hy * hx;
            wt[s][1] = vmul * hy * lx;
            wt[s][2] = vmul * ly * hx;
            wt[s][3] = vmul * ly * lx;
        }

        const size_t plane = (size_t)H * (size_t)H;
        const float* bp = feats[l] + ((size_t)bIdx * CTOT + c0) * plane;

#pragma unroll
        for (int cc = 0; cc < CCHUNK; ++cc) {
            const float* p = bp + (size_t)cc * plane;
            if (cc + 1 < CCHUNK)
                __builtin_prefetch(p + plane + off[0], 0, 0);   // global_prefetch_b8
            float v = 0.0f;
#pragma unroll
            for (int s = 0; s < 4; ++s) {
                const float* r0 = p + off[s];
                const float* r1 = r0 + H;
                v += wt[s][0] * r0[0] + wt[s][1] * r0[1]
                   + wt[s][2] * r1[0] + wt[s][3] * r1[1];
            }
            acc[cc] = fmaxf(acc[cc], v);
        }
    }

    // Output is stream-out (write-once): non-temporal stores keep the 178MB
    // feature pyramid resident in the 192MB L2 instead of cycling 128MB of
    // dirty output lines through it.
    float* op = out + ((size_t)n * CTOT + c0) * NBINS + tid;
#pragma unroll
    for (int cc = 0; cc < CCHUNK; ++cc)
        __builtin_nontemporal_store(acc[cc], op + (size_t)cc * NBINS);
}

extern "C" void kernel_launch(void* const* d_in, const int* in_sizes, int n_in,
                              void* d_out, int out_size, void* d_ws, size_t ws_size,
                              hipStream_t stream) {
    (void)n_in; (void)out_size; (void)d_ws; (void)ws_size;
    const float* f0   = (const float*)d_in[0];
    const float* f1   = (const float*)d_in[1];
    const float* f2   = (const float*)d_in[2];
    const float* f3   = (const float*)d_in[3];
    const float* rois = (const float*)d_in[4];
    float* out = (float*)d_out;

    const int N = in_sizes[4] / 6;   // 512
    dim3 block(OUT_W, OUT_H, 1);     // 245 threads = 8 wave32
    dim3 grid(N, CTOT / CCHUNK, 1);  // 512 x 16 blocks
    rroi_align_max_kernel<<<grid, block, 0, stream>>>(f0, f1, f2, f3, rois, out);
}